// _PCL_Losses_57964878627194
// MI455X (gfx1250) — compile-verified
//
#include <hip/hip_runtime.h>

#define EPS_F 1e-6f
#define N_PROP 262144
#define N_PC 4096
#define N_CLS 81
#define BLOCK 256
#define NBLK_MAX 1024

typedef float v2f __attribute__((ext_vector_type(2)));
typedef float v8f __attribute__((ext_vector_type(8)));

// Wave32 all-lane sum via the matrix pipe: D = A x ones(4x16) + 0 with
// V_WMMA_F32_16X16X4_F32. Each lane contributes its partial through the A
// matrix (lanes 0-15 land in K=0, lanes 16-31 in K=2 of the 16x4 A layout),
// so D[m,n] = local(m) + local(m+16) replicated across n. Summing the 8 D
// VGPRs per lane gives the two half-wave sums; one xor-16 shuffle merges them.
// Exact f32 math, EXEC must be all ones at the call site.
__device__ __forceinline__ float wave_sum_wmma(float v) {
    v2f a; a.x = v;    a.y = 0.0f;
    v2f b; b.x = 1.0f; b.y = 1.0f;
    v8f c = {};
    c = __builtin_amdgcn_wmma_f32_16x16x4_f32(
        /*neg_a=*/false, a, /*neg_b=*/false, b,
        /*c_mod=*/(short)0, c, /*reuse_a=*/false, /*reuse_b=*/false);
    float s = (((c[0] + c[1]) + (c[2] + c[3])) + ((c[4] + c[5]) + (c[6] + c[7])));
    s += __shfl_xor(s, 16, 32);
    return s;   // full 32-lane sum in every lane
}

// Deterministic block reduction: WMMA wave sums -> LDS -> fixed-order scan by
// thread 0. Result valid on thread 0 only.
__device__ __forceinline__ float block_sum(float v, float* lds) {
    float ws = wave_sum_wmma(v);
    const int lane = threadIdx.x & 31;
    const int wave = threadIdx.x >> 5;
    if (lane == 0) lds[wave] = ws;
    __syncthreads();
    float total = 0.0f;
    if (threadIdx.x == 0) {
        const int nw = BLOCK >> 5;
        #pragma unroll
        for (int i = 0; i < nw; ++i) total += lds[i];
    }
    return total;
}

// Pass 1: background term. One partial per block (deterministic within block).
__global__ void __launch_bounds__(BLOCK)
pcl_bg_partial(const float* __restrict__ prob,      // [N_PROP, N_CLS], col 0 used
               const int*   __restrict__ labels,    // [N_PROP]
               const float* __restrict__ w,         // [N_PROP]
               const float* __restrict__ im_labels, // [N_CLS]
               float*       __restrict__ partial)   // [gridDim.x]
{
    __shared__ float lds[BLOCK / 32];
    const float present0 = (im_labels[0] > 0.0f) ? 1.0f : 0.0f;

    float t = 0.0f;
    const int stride = gridDim.x * BLOCK;
    for (int i = blockIdx.x * BLOCK + threadIdx.x; i < N_PROP; i += stride) {
        if (labels[i] == 0) {
            const float p = prob[(size_t)i * N_CLS];   // strided column-0 gather
            t += w[i] * __logf(fmaxf(p, EPS_F));
        }
    }
    t *= present0;

    const float total = block_sum(t, lds);
    if (threadIdx.x == 0) partial[blockIdx.x] = total;
}

// Pass 2: foreground term + combine partials + finalize. Single block,
// fixed-order traversal -> bit-deterministic.
__global__ void __launch_bounds__(BLOCK)
pcl_final(const float* __restrict__ partial,    // [nblk]
          int                       nblk,
          const int*   __restrict__ pc_labels,  // [N_PC]
          const float* __restrict__ pc_probs,   // [N_PC]
          const float* __restrict__ pc_count,   // [N_PC]
          const float* __restrict__ img_w,      // [N_PC]
          const float* __restrict__ im_labels,  // [N_CLS]
          float*       __restrict__ out)        // [1]
{
    __shared__ float lds[BLOCK / 32];

    float s = 0.0f;
    for (int j = threadIdx.x; j < nblk; j += BLOCK)
        s += partial[j];

    for (int p = threadIdx.x; p < N_PC; p += BLOCK) {
        const int c = pc_labels[p];
        if (c > 0 && im_labels[c] > 0.0f)
            s += pc_count[p] * img_w[p] * __logf(fmaxf(pc_probs[p], EPS_F));
    }

    const float total = block_sum(s, lds);
    if (threadIdx.x == 0)
        out[0] = -total / (float)N_PROP;
}

extern "C" void kernel_launch(void* const* d_in, const int* in_sizes, int n_in,
                              void* d_out, int out_size, void* d_ws, size_t ws_size,
                              hipStream_t stream) {
    // setup_inputs() order:
    const float* prob      = (const float*)d_in[0]; // pcl_prob [N_PROP*N_CLS]
    const int*   labels    = (const int*)  d_in[1]; // labels [N_PROP]
    const float* w         = (const float*)d_in[2]; // cls_loss_weights [N_PROP]
    // d_in[3] = gt_assignment (unused by the reference)
    const int*   pc_labels = (const int*)  d_in[4]; // [N_PC]
    const float* pc_probs  = (const float*)d_in[5]; // [N_PC]
    const float* pc_count  = (const float*)d_in[6]; // [N_PC]
    const float* img_w     = (const float*)d_in[7]; // [N_PC]
    const float* im_labels = (const float*)d_in[8]; // [N_CLS]

    float* partial = (float*)d_ws;
    int nblk = NBLK_MAX;
    const size_t cap = ws_size / sizeof(float);
    if ((size_t)nblk > cap) nblk = (int)cap;
    if (nblk < 1) nblk = 1;

    pcl_bg_partial<<<nblk, BLOCK, 0, stream>>>(prob, labels, w, im_labels, partial);
    pcl_final<<<1, BLOCK, 0, stream>>>(partial, nblk, pc_labels, pc_probs,
                                       pc_count, img_w, im_labels, (float*)d_out);
}